// GNN_5823975653592
// MI455X (gfx1250) — compile-verified
//
#include <hip/hip_runtime.h>
#include <cstdint>

typedef float v2f __attribute__((ext_vector_type(2)));
typedef float v8f __attribute__((ext_vector_type(8)));

#define F_IN 288
#define H1   16
#define H2   32
#define F_OUT 7

// ---------- degree / norm ----------
__global__ void init_deg_kernel(float* __restrict__ deg, int N) {
    int i = blockIdx.x * blockDim.x + threadIdx.x;
    if (i < N) deg[i] = 1.0f;  // self-loop contributes 1 to every node's degree
}

__global__ void zero_kernel(float* __restrict__ p, int n) {
    int i = blockIdx.x * blockDim.x + threadIdx.x;
    if (i < n) p[i] = 0.0f;
}

__global__ void deg_kernel(const long long* __restrict__ dst, float* __restrict__ deg, int E) {
    int e = blockIdx.x * blockDim.x + threadIdx.x;
    if (e < E) atomicAdd(&deg[(int)dst[e]], 1.0f);
}

__global__ void dinv_kernel(float* __restrict__ deg, int N) {
    int i = blockIdx.x * blockDim.x + threadIdx.x;
    if (i < N) deg[i] = rsqrtf(deg[i]);   // deg >= 1 always (self-loops)
}

// ---------- GEMM1: t1[N,16] = x[N,288] @ W1[288,16] via V_WMMA_F32_16X16X4_F32 ----------
// One wave per 16-row tile. A-frag (16x4 f32): lanes 0-15 hold M=lane, K={k,k+1};
// lanes 16-31 hold M=lane-16, K={k+2,k+3}. B-frag (4x16) mirrored with N=lane&15.
__global__ void gemm1_wmma(const float* __restrict__ x, const float* __restrict__ W,
                           float* __restrict__ t1, int ntiles) {
    int wave = (int)((blockIdx.x * blockDim.x + threadIdx.x) >> 5);
    int lane = threadIdx.x & 31;
    if (wave >= ntiles) return;
    int r  = lane & 15;          // M index (A) / N index (B,C)
    int kb = (lane >> 4) << 1;   // 0 or 2
    const float* xrow = x + (size_t)(wave * 16 + r) * F_IN;
    v8f c = {};
#pragma unroll 4
    for (int k = 0; k < F_IN; k += 4) {
        v2f a, b;
        a.x = xrow[k + kb];
        a.y = xrow[k + kb + 1];
        b.x = W[(k + kb)     * H1 + r];
        b.y = W[(k + kb + 1) * H1 + r];
        c = __builtin_amdgcn_wmma_f32_16x16x4_f32(false, a, false, b, (short)0, c, false, false);
    }
    // C layout: vgpr j, lane L -> C[(L>>4)*8 + j][L&15]
    int mb = (lane >> 4) * 8;
    float* o = t1 + (size_t)(wave * 16 + mb) * H1 + r;
#pragma unroll
    for (int j = 0; j < 8; ++j) o[(size_t)j * H1] = c[j];
}

// ---------- GEMM2: t2[N,32] = h1[N,16] @ W2[16,32] ----------
__global__ void gemm2_wmma(const float* __restrict__ h1, const float* __restrict__ W,
                           float* __restrict__ t2, int ntiles) {
    int wave = (int)((blockIdx.x * blockDim.x + threadIdx.x) >> 5);
    int lane = threadIdx.x & 31;
    if (wave >= ntiles) return;
    int r  = lane & 15;
    int kb = (lane >> 4) << 1;
    const float* hrow = h1 + (size_t)(wave * 16 + r) * H1;
    v8f c0 = {}, c1 = {};
#pragma unroll
    for (int k = 0; k < H1; k += 4) {
        v2f a, b0, b1;
        a.x  = hrow[k + kb];
        a.y  = hrow[k + kb + 1];
        b0.x = W[(k + kb)     * H2 + r];
        b0.y = W[(k + kb + 1) * H2 + r];
        b1.x = W[(k + kb)     * H2 + 16 + r];
        b1.y = W[(k + kb + 1) * H2 + 16 + r];
        c0 = __builtin_amdgcn_wmma_f32_16x16x4_f32(false, a, false, b0, (short)0, c0, false, false);
        c1 = __builtin_amdgcn_wmma_f32_16x16x4_f32(false, a, false, b1, (short)0, c1, false, false);
    }
    int mb = (lane >> 4) * 8;
    float* o = t2 + (size_t)(wave * 16 + mb) * H2;
#pragma unroll
    for (int j = 0; j < 8; ++j) {
        o[(size_t)j * H2 + r]      = c0[j];
        o[(size_t)j * H2 + 16 + r] = c1[j];
    }
}

// ---------- edge aggregation: h[dst] += t[src] * dinv[src]*dinv[dst] ----------
template <int F, int SHIFT>
__global__ void agg_kernel(const long long* __restrict__ ei, const float* __restrict__ dinv,
                           const float* __restrict__ t, float* __restrict__ h, int E) {
    long long tid = (long long)blockIdx.x * blockDim.x + threadIdx.x;
    int e = (int)(tid >> SHIFT);
    if (e >= E) return;
    int f = (int)tid & (F - 1);
    int s = (int)ei[e];
    int d = (int)ei[(size_t)E + e];
    float w = dinv[s] * dinv[d];
    atomicAdd(&h[(size_t)d * F + f], t[(size_t)s * F + f] * w);
}

// ---------- self-loop + bias + relu (in place on h) ----------
template <int F, int SHIFT>
__global__ void relu_bias_kernel(float* __restrict__ h, const float* __restrict__ t,
                                 const float* __restrict__ dinv, const float* __restrict__ b, int N) {
    int tid = blockIdx.x * blockDim.x + threadIdx.x;
    int i = tid >> SHIFT;
    if (i >= N) return;
    int f = tid & (F - 1);
    float di = dinv[i];
    size_t idx = (size_t)i * F + f;
    float v = h[idx] + t[idx] * (di * di) + b[f];
    h[idx] = v > 0.0f ? v : 0.0f;
}

// ---------- final: out = relu(h2agg + t2*dinv^2 + b2) @ Wfc + bfc ----------
__global__ void final_kernel(const float* __restrict__ h2agg, const float* __restrict__ t2,
                             const float* __restrict__ dinv, const float* __restrict__ b2,
                             const float* __restrict__ Wfc, const float* __restrict__ bfc,
                             float* __restrict__ out, int N) {
    int i = blockIdx.x * blockDim.x + threadIdx.x;
    if (i >= N) return;
    float di = dinv[i];
    float sw = di * di;
    const float4* ha = (const float4*)(h2agg + (size_t)i * H2);
    const float4* ta = (const float4*)(t2    + (size_t)i * H2);
    float h[H2];
#pragma unroll
    for (int q = 0; q < H2 / 4; ++q) {
        float4 av = ha[q];
        float4 tv = ta[q];
        float v0 = av.x + tv.x * sw + b2[4 * q + 0];
        float v1 = av.y + tv.y * sw + b2[4 * q + 1];
        float v2 = av.z + tv.z * sw + b2[4 * q + 2];
        float v3 = av.w + tv.w * sw + b2[4 * q + 3];
        h[4 * q + 0] = v0 > 0.f ? v0 : 0.f;
        h[4 * q + 1] = v1 > 0.f ? v1 : 0.f;
        h[4 * q + 2] = v2 > 0.f ? v2 : 0.f;
        h[4 * q + 3] = v3 > 0.f ? v3 : 0.f;
    }
    float acc[F_OUT];
#pragma unroll
    for (int o = 0; o < F_OUT; ++o) acc[o] = bfc[o];
#pragma unroll
    for (int f = 0; f < H2; ++f) {
        float hf = h[f];
#pragma unroll
        for (int o = 0; o < F_OUT; ++o) acc[o] += hf * Wfc[f * F_OUT + o];
    }
    float* op = out + (size_t)i * F_OUT;
#pragma unroll
    for (int o = 0; o < F_OUT; ++o) op[o] = acc[o];
}

extern "C" void kernel_launch(void* const* d_in, const int* in_sizes, int n_in,
                              void* d_out, int out_size, void* d_ws, size_t ws_size,
                              hipStream_t stream) {
    const float*     x   = (const float*)d_in[0];
    const long long* ei  = (const long long*)d_in[1];
    const float*     W1  = (const float*)d_in[2];
    const float*     b1  = (const float*)d_in[3];
    const float*     W2  = (const float*)d_in[4];
    const float*     b2  = (const float*)d_in[5];
    const float*     Wfc = (const float*)d_in[6];
    const float*     bfc = (const float*)d_in[7];
    float*           out = (float*)d_out;

    const int N = in_sizes[0] / F_IN;  // 100000
    const int E = in_sizes[1] / 2;     // 3200000

    // workspace layout (floats); N is a multiple of 16 so everything stays 16B-aligned
    float* ws   = (float*)d_ws;
    float* dinv = ws;
    float* t1   = dinv + (size_t)N;
    float* h1   = t1   + (size_t)N * H1;
    float* t2   = h1   + (size_t)N * H1;
    float* h2   = t2   + (size_t)N * H2;

    const int ntiles = (N + 15) / 16;  // 6250
    const int TB = 256;

    // degrees -> dinv
    init_deg_kernel<<<(N + TB - 1) / TB, TB, 0, stream>>>(dinv, N);
    deg_kernel<<<(E + TB - 1) / TB, TB, 0, stream>>>(ei + E, dinv, E);
    dinv_kernel<<<(N + TB - 1) / TB, TB, 0, stream>>>(dinv, N);

    // layer 1
    gemm1_wmma<<<(ntiles + 1) / 2, 64, 0, stream>>>(x, W1, t1, ntiles);
    zero_kernel<<<((N * H1) + TB - 1) / TB, TB, 0, stream>>>(h1, N * H1);
    {
        long long total = (long long)E * H1;
        agg_kernel<H1, 4><<<(unsigned)((total + TB - 1) / TB), TB, 0, stream>>>(ei, dinv, t1, h1, E);
    }
    relu_bias_kernel<H1, 4><<<((N * H1) + TB - 1) / TB, TB, 0, stream>>>(h1, t1, dinv, b1, N);

    // layer 2
    gemm2_wmma<<<(ntiles + 1) / 2, 64, 0, stream>>>(h1, W2, t2, ntiles);
    zero_kernel<<<((N * H2) + TB - 1) / TB, TB, 0, stream>>>(h2, N * H2);
    {
        long long total = (long long)E * H2;
        agg_kernel<H2, 5><<<(unsigned)((total + TB - 1) / TB), TB, 0, stream>>>(ei, dinv, t2, h2, E);
    }

    // fused relu+bias+linear head
    final_kernel<<<(N + TB - 1) / TB, TB, 0, stream>>>(h2, t2, dinv, b2, Wfc, bfc, out, N);
}